// PosMix_28028956574046
// MI455X (gfx1250) — compile-verified
//
#include <hip/hip_runtime.h>

#define LMAX   13
#define KPAD   16      // K padded to 16 (4 wmma k-steps of 4)
#define KP     20      // Wlds/Bl physical row stride (floats): 8B align, no bank conflicts
#define DP     52      // Dst per-batch row stride (floats): 48 data + pad, multiple of 4
#define NTYPES 20
#define DV     64
#define NB     16      // batch
#define NCOL   48      // NB*3 columns (b,d)
#define WAVES  8
#define TPB    256

__device__ __constant__ int RES_LEN[NTYPES] = {3,4,5,5,6,6,6,7,7,7,7,7,8,8,8,9,10,10,11,13};

typedef float v2f __attribute__((ext_vector_type(2)));
typedef float v4f __attribute__((ext_vector_type(4)));
typedef float v8f __attribute__((ext_vector_type(8)));

// ---------------------------------------------------------------------------
// Prefix scan of residue lengths -> flat atom start offsets (runs once, 1 block)
// ---------------------------------------------------------------------------
__global__ void posmix_scan_starts(const int* __restrict__ seq,
                                   int* __restrict__ starts, int nres) {
    __shared__ int part[TPB];
    const int t = threadIdx.x;
    const int chunk = (nres + TPB - 1) / TPB;
    const int lo = t * chunk;
    const int hi = (lo + chunk < nres) ? (lo + chunk) : nres;
    int s = 0;
    for (int j = lo; j < hi; ++j) s += RES_LEN[seq[j]];
    part[t] = s;
    __syncthreads();
    if (t == 0) {                       // serial exclusive scan of 256 partials
        int acc = 0;
        for (int k = 0; k < TPB; ++k) { int v = part[k]; part[k] = acc; acc += v; }
    }
    __syncthreads();
    int acc = part[t];
    for (int j = lo; j < hi; ++j) { starts[j] = acc; acc += RES_LEN[seq[j]]; }
}

// ---------------------------------------------------------------------------
// Main kernel: one workgroup (8 wave32) per residue.
//   Stacked GEMM:  [ W_amn[t] (64 x L) ; W_atm[t] (L*64 x L) ]  x  diff(L x 48)
//   via V_WMMA_F32_16X16X4_F32, K padded 13->16, N = 48 cols = (b,d).
//   Accumulators are scattered into output-ordered LDS rows, then streamed
//   out with aligned non-temporal global_store_b128.
// ---------------------------------------------------------------------------
__global__ void __launch_bounds__(TPB)
posmix_wmma_kernel(const float* __restrict__ pos_atm,
                   const float* __restrict__ pos_amn,
                   const float* __restrict__ W_amn,
                   const float* __restrict__ W_atm,
                   const int* __restrict__ seq,
                   const int* __restrict__ starts,
                   float* __restrict__ out_atm,
                   float* __restrict__ out_amn,
                   int atoms, int nres)
{
    __shared__ float Wlds[(DV + LMAX * DV) * KP];   // up to 896 rows x 20
    __shared__ float Bl[NCOL * KP];                 // diff, [n][k]
    __shared__ float Dst[WAVES * NB * DP];          // per-wave staging, output order [b][e]

    const int i    = blockIdx.x;
    const int t    = seq[i];
    const int L    = RES_LEN[t];
    const int s0   = starts[i];
    const int tid  = threadIdx.x;
    const int lane = tid & 31;
    const int w    = tid >> 5;

    const int Mrows  = DV + L * DV;     // stacked weight rows (multiple of 16)
    const int Mtiles = Mrows >> 4;      // 4*(L+1) <= 56

    // ---- load stacked weights into LDS (rows: [0,64)=W_amn[t], [64,..)=W_atm[t])
    const float* wa = W_amn + (size_t)t * DV * LMAX;
    const float* wt = W_atm + (size_t)t * LMAX * DV * LMAX;
    for (int idx = tid; idx < Mrows * KPAD; idx += TPB) {
        const int row = idx >> 4;
        const int k   = idx & 15;
        float v = 0.0f;
        if (k < LMAX)  // l >= L entries are annihilated by zero-padded B matrix
            v = (row < DV) ? wa[row * LMAX + k] : wt[(row - DV) * LMAX + k];
        Wlds[row * KP + k] = v;
    }

    // ---- build B matrix: Bl[n][k] = pos_atm[b, s0+k, d] - pos_amn[b, i, d]
    for (int idx = tid; idx < NCOL * KPAD; idx += TPB) {
        const int n = idx >> 4;
        const int k = idx & 15;
        const int b = n / 3;
        const int d = n - b * 3;
        float v = 0.0f;
        if (k < L)
            v = pos_atm[((size_t)b * atoms + s0 + k) * 3 + d]
              - pos_amn[((size_t)b * nres  + i)      * 3 + d];
        Bl[n * KP + k] = v;
    }
    __syncthreads();

    const int half = lane >> 4;     // 0: K pair (k0,k0+1)   1: (k0+2,k0+3)
    const int l16  = lane & 15;
    const int lrb  = half << 3;     // D-fragment local row base (0 or 8)
    float* dst = Dst + w * (NB * DP);

    // ---- per-lane staging scatter bases: acc group g holds col n=16g+l16=(b,d),
    //      rows lrb..lrb+7; output element e = lrow*3 + d in row b.
    int sbase[3];
    #pragma unroll
    for (int g = 0; g < 3; ++g) {
        const int n  = 16 * g + l16;
        const int bc = n / 3;
        const int dc = n - 3 * bc;
        sbase[g] = bc * DP + dc + lrb * 3;
    }

    // ---- per-lane readout phases: 768 strip floats = 192 aligned v4 chunks
    int roLds[6], roB[6], roE[6];
    #pragma unroll
    for (int q = 0; q < 6; ++q) {
        const int c  = lane + 32 * q;   // chunk id 0..191
        const int bq = c / 12;          // batch row
        const int qq = c - 12 * bq;     // v4 chunk within 48-float row
        roLds[q] = bq * DP + 4 * qq;
        roB[q]   = bq;
        roE[q]   = 4 * qq;
    }

    for (int mt = w; mt < Mtiles; mt += WAVES) {
        const int m0 = mt << 4;
        v8f acc0 = {}, acc1 = {}, acc2 = {};
        #pragma unroll
        for (int ks = 0; ks < 4; ++ks) {
            const int kb = (ks << 2) + (half << 1);
            const v2f a  = *(const v2f*)&Wlds[(m0 + l16) * KP + kb];
            const v2f b0 = *(const v2f*)&Bl[( 0 + l16) * KP + kb];
            const v2f b1 = *(const v2f*)&Bl[(16 + l16) * KP + kb];
            const v2f b2 = *(const v2f*)&Bl[(32 + l16) * KP + kb];
            acc0 = __builtin_amdgcn_wmma_f32_16x16x4_f32(false, a, false, b0, (short)0, acc0, false, false);
            acc1 = __builtin_amdgcn_wmma_f32_16x16x4_f32(false, a, false, b1, (short)0, acc1, false, false);
            acc2 = __builtin_amdgcn_wmma_f32_16x16x4_f32(false, a, false, b2, (short)0, acc2, false, false);
        }

        // ---- scatter accumulators into output-ordered staging (imm-offset b32s)
        #pragma unroll
        for (int r = 0; r < 8; ++r) dst[sbase[0] + 3 * r] = acc0[r];
        #pragma unroll
        for (int r = 0; r < 8; ++r) dst[sbase[1] + 3 * r] = acc1[r];
        #pragma unroll
        for (int r = 0; r < 8; ++r) dst[sbase[2] + 3 * r] = acc2[r];

        // ---- strip output base (64B-aligned: vb multiple of 16 -> *3 mult of 48)
        float* bptr;
        size_t bstride;
        if (mt < 4) {                                   // x_v_amn rows o = m0..m0+15
            bptr    = out_amn + ((size_t)i * DV + m0) * 3;
            bstride = (size_t)nres * DV * 3;
        } else {                                        // x_v_atm atom slot mm, v base vb
            const int mm = (mt - 4) >> 2;
            const int vb = ((mt - 4) & 3) << 4;
            bptr    = out_atm + ((size_t)(s0 + mm) * DV + vb) * 3;
            bstride = (size_t)atoms * DV * 3;
        }

        // ---- wide coalesced non-temporal writeout: 6 x global_store_b128/lane
        #pragma unroll
        for (int q = 0; q < 6; ++q) {
            const v4f v = *(const v4f*)&dst[roLds[q]];
            __builtin_nontemporal_store(v, (v4f*)(bptr + (size_t)roB[q] * bstride + roE[q]));
        }
    }
}

// ---------------------------------------------------------------------------
extern "C" void kernel_launch(void* const* d_in, const int* in_sizes, int n_in,
                              void* d_out, int out_size, void* d_ws, size_t ws_size,
                              hipStream_t stream) {
    (void)n_in; (void)out_size; (void)ws_size;
    const float* pos_atm = (const float*)d_in[0];
    const float* pos_amn = (const float*)d_in[1];
    const float* W_amn   = (const float*)d_in[2];
    const float* W_atm   = (const float*)d_in[3];
    const int*   seq     = (const int*)d_in[4];

    const int nres  = in_sizes[4];
    const int atoms = in_sizes[0] / (NB * 3);

    float* out_atm = (float*)d_out;                                  // (B, atoms, 64, 3)
    float* out_amn = out_atm + (size_t)NB * atoms * DV * 3;          // (B, nres, 64, 3)
    int*   starts  = (int*)d_ws;

    posmix_scan_starts<<<1, TPB, 0, stream>>>(seq, starts, nres);
    posmix_wmma_kernel<<<nres, TPB, 0, stream>>>(pos_atm, pos_amn, W_amn, W_atm,
                                                 seq, starts, out_atm, out_amn,
                                                 atoms, nres);
}